// QuantumLayer_28432683499684
// MI455X (gfx1250) — compile-verified
//
#include <hip/hip_runtime.h>
#include <math.h>

// out[b] = cospi(x[b,1]) * cospi(x[b,3]) * cos(theta[1]) * cos(theta[3])
// (analytic reduction of the 4-qubit RX->RY->RZ->CNOT parity-expectation circuit)

#define TILE_ROWS 256

__global__ __launch_bounds__(256) void qparity_kernel(
    const float* __restrict__ inputs,   // [B,4] row-major
    const float* __restrict__ theta,    // [8]
    float* __restrict__ out,            // [B]
    int B)
{
    __shared__ __align__(16) float smem[TILE_ROWS * 4];   // 4 KB tile

    const int t = threadIdx.x;
    const long long row = (long long)blockIdx.x * TILE_ROWS + t;
    const bool active = row < (long long)B;

    // CDNA5 async tensor-data path: stage this block's input tile into LDS.
    // GVS addressing: uniform SGPR base (block tile start) + per-lane VGPR
    // byte offset; VDST vgpr carries the LDS byte address.
    const float* gbase = inputs + (long long)blockIdx.x * (TILE_ROWS * 4);
    unsigned voff    = (unsigned)(t * 16);                        // 16 B per row
    unsigned lds_off = (unsigned)(unsigned long long)(&smem[t * 4]); // low 32 bits = LDS offset

    if (active) {
        asm volatile("global_load_async_to_lds_b128 %0, %1, %2"
                     :
                     : "v"(lds_off), "v"(voff), "s"(gbase)
                     : "memory");
    }
    // Wait for this wave's async transfers, then block barrier so every
    // wave's LDS writes are visible.
    asm volatile("s_wait_asynccnt 0" ::: "memory");
    __syncthreads();

    if (active) {
        float4 v = *(const float4*)(&smem[t * 4]);   // ds_load_b128

        // Uniform scalar factor (s_load + 2x cos, negligible).
        float K = cosf(theta[1]) * cosf(theta[3]);

        float r = cospif(v.y) * cospif(v.w) * K;

        // Streamed write-once output: non-temporal store.
        __builtin_nontemporal_store(r, out + row);
    }
}

extern "C" void kernel_launch(void* const* d_in, const int* in_sizes, int n_in,
                              void* d_out, int out_size, void* d_ws, size_t ws_size,
                              hipStream_t stream) {
    const float* inputs = (const float*)d_in[0];   // [B,4] float32
    const float* theta  = (const float*)d_in[1];   // [8]   float32
    float* out = (float*)d_out;                    // [B]   float32

    const int B = in_sizes[0] / 4;
    const int blocks = (B + TILE_ROWS - 1) / TILE_ROWS;

    hipLaunchKernelGGL(qparity_kernel, dim3(blocks), dim3(TILE_ROWS), 0, stream,
                       inputs, theta, out, B);
}